// AdvanceGNNModel_26164940767513
// MI455X (gfx1250) — compile-verified
//
#include <hip/hip_runtime.h>
#include <hip/hip_bf16.h>

// ---------------------------------------------------------------------------
// GCN on MI455X (gfx1250). Dense transforms use V_WMMA_F32_16X16X4_F32 (f32
// matrix pipe, bit-compatible precision with the f32 reference). Aggregation
// is L2-resident f32 atomics with column-coalesced lanes. Leaky-ReLU is fused
// into the next layer's GEMM A-load (compile-time templated, no per-element
// flag selects); edge norms are precomputed once.
// ---------------------------------------------------------------------------

typedef __attribute__((ext_vector_type(2))) float v2f;
typedef __attribute__((ext_vector_type(8))) float v8f;

#define DIN  128
#define DH   64
#define DOUT 4

__device__ __forceinline__ float leaky(float v) {
    return v >= 0.f ? v : 0.2f * v;
}

// ---- degree / norm precompute ---------------------------------------------
__global__ void k_set_ones(float* p, int n) {
    int i = blockIdx.x * blockDim.x + threadIdx.x;
    if (i < n) p[i] = 1.0f;  // self-loop contributes 1 to every degree
}

__global__ void k_accum_deg(const int* __restrict__ dst, float* deg, int e) {
    int i = blockIdx.x * blockDim.x + threadIdx.x;
    if (i < e) atomicAdd(&deg[dst[i]], 1.0f);
}

__global__ void k_rsqrt(float* p, int n) {
    int i = blockIdx.x * blockDim.x + threadIdx.x;
    if (i < n) p[i] = rsqrtf(p[i]);  // deg >= 1 always (self-loops)
}

// wnorm[e] = dinv[src[e]] * dinv[dst[e]]  (one-time, reused by all 6 layers)
__global__ void k_edge_norm(const int* __restrict__ src, const int* __restrict__ dst,
                            const float* __restrict__ dinv, float* __restrict__ wnorm,
                            int e) {
    int i = blockIdx.x * blockDim.x + threadIdx.x;
    if (i < e) wnorm[i] = dinv[src[i]] * dinv[dst[i]];
}

// ---- dense transform: H[N x 64] = act(X)[N x K] @ W[K x 64] via WMMA f32 --
// 4 waves per block; wave w computes the 16x16 tile at cols [16w, 16w+16).
// f32 A (16x4) layout : lanes 0-15 hold K={k0,k0+1} in v0,v1 for row M=lane;
//                       lanes 16-31 hold K={k0+2,k0+3} for row M=lane-16.
// f32 B (4x16) layout : mirrored (row K striped across lanes, col = lane&15).
// f32 C/D (16x16)     : vgpr r holds row M = r + 8*(lane>>4), col N = lane&15.
template <int ACT>
__global__ __launch_bounds__(128)
void k_gemm_wmma_f32(const float* __restrict__ X, const float* __restrict__ W,
                     float* __restrict__ H, int K) {
    const int lane = threadIdx.x & 31;
    const int wid  = threadIdx.x >> 5;   // col tile 0..3
    const int half = lane >> 4;          // 0 or 1
    const int mn   = lane & 15;          // row for A, col for B/D
    const int row0 = blockIdx.x * 16;
    const int col0 = wid * 16;
    const int NC   = DH;

    const float* xrow = X + (size_t)(row0 + mn) * K;

    v8f c = {0.f, 0.f, 0.f, 0.f, 0.f, 0.f, 0.f, 0.f};
    for (int k0 = 0; k0 < K; k0 += 4) {
        const int ka = k0 + 2 * half;    // even -> 8B-aligned b64 load
        v2f a = *(const v2f*)(xrow + ka);
        if (ACT) { a.x = leaky(a.x); a.y = leaky(a.y); }  // fused activation
        v2f b;
        b.x = W[(size_t)ka       * NC + col0 + mn];
        b.y = W[(size_t)(ka + 1) * NC + col0 + mn];
        // 8 args: (neg_a, A, neg_b, B, c_mod, C, reuse_a, reuse_b)
        c = __builtin_amdgcn_wmma_f32_16x16x4_f32(false, a, false, b,
                                                  (short)0, c, false, false);
    }
#pragma unroll
    for (int r = 0; r < 8; ++r) {
        H[(size_t)(row0 + r + 8 * half) * NC + col0 + mn] = c[r];
    }
}

// ---- layer 5 transform (dout=4, too narrow for a 16-wide WMMA tile) -------
__global__ void k_gemm_small(const float* __restrict__ X, const float* __restrict__ W,
                             float* __restrict__ H, int n, int K, int M) {
    int idx = blockIdx.x * blockDim.x + threadIdx.x;
    if (idx >= n * M) return;
    int row = idx / M, col = idx % M;
    const float* xr = X + (size_t)row * K;
    float acc = 0.f;
    for (int k = 0; k < K; ++k) acc = fmaf(leaky(xr[k]), W[(size_t)k * M + col], acc);
    H[idx] = acc;
}

// ---- agg init: self-loop contribution + bias ------------------------------
// agg[n, c] = h[n, c] * dinv[n]^2 + b[c]
__global__ void k_init_agg(const float* __restrict__ h, const float* __restrict__ dinv,
                           const float* __restrict__ b, float* __restrict__ agg,
                           int n, int dout) {
    int idx = blockIdx.x * blockDim.x + threadIdx.x;
    if (idx >= n * dout) return;
    int node = idx / dout, c = idx % dout;
    float dv = dinv[node];
    agg[idx] = h[idx] * dv * dv + b[c];
}

// ---- edge scatter: agg[dst] += h[src] * wnorm[e] --------------------------
// Consecutive lanes take consecutive columns of one edge -> coalesced atomics.
__global__ void k_edge_scatter(const int* __restrict__ src, const int* __restrict__ dst,
                               const float* __restrict__ wnorm, const float* __restrict__ h,
                               float* __restrict__ agg, int e, int dout) {
    const int epb = 256 / dout;  // edges per 256-thread block
    int t = threadIdx.x;
    int edge = blockIdx.x * epb + t / dout;
    int c = t % dout;
    if (edge >= e) return;
    int s = src[edge], d = dst[edge];
    float w = wnorm[edge];
    atomicAdd(&agg[(size_t)d * dout + c], h[(size_t)s * dout + c] * w);
}

static inline int cdiv(long long a, long long b) { return (int)((a + b - 1) / b); }

extern "C" void kernel_launch(void* const* d_in, const int* in_sizes, int n_in,
                              void* d_out, int out_size, void* d_ws, size_t ws_size,
                              hipStream_t stream) {
    const float* x  = (const float*)d_in[0];
    const int*   ei = (const int*)d_in[1];
    const int N = in_sizes[0] / DIN;
    const int E = in_sizes[1] / 2;
    const int* src = ei;       // edge_index[0] = message sources
    const int* dst = ei + E;   // edge_index[1] = aggregation targets

    const float* Ws[6], * bs[6];
    for (int i = 0; i < 6; ++i) { Ws[i] = (const float*)d_in[2 + 2 * i];
                                  bs[i] = (const float*)d_in[3 + 2 * i]; }

    // workspace (floats): dinv[N] | wnorm[E] | h[N*DH] | aggA[N*DH] | aggB[N*DH]
    float* dinv  = (float*)d_ws;
    float* wnorm = dinv  + N;
    float* hbuf  = wnorm + E;
    float* aggA  = hbuf  + (size_t)N * DH;
    float* aggB  = aggA  + (size_t)N * DH;

    // --- graph norm: deg -> rsqrt -> per-edge weight ---
    k_set_ones  <<<cdiv(N, 256), 256, 0, stream>>>(dinv, N);
    k_accum_deg <<<cdiv(E, 256), 256, 0, stream>>>(dst, dinv, E);
    k_rsqrt     <<<cdiv(N, 256), 256, 0, stream>>>(dinv, N);
    k_edge_norm <<<cdiv(E, 256), 256, 0, stream>>>(src, dst, dinv, wnorm, E);

    // --- layers 0..4 (dout = 64); leaky-ReLU fused into next layer's A-load ---
    float* aggCur = aggA;
    float* aggNxt = aggB;
    const float* in = x;

    for (int layer = 0; layer < 5; ++layer) {
        if (layer == 0)
            k_gemm_wmma_f32<0><<<N / 16, 128, 0, stream>>>(in, Ws[layer], hbuf, DIN);
        else
            k_gemm_wmma_f32<1><<<N / 16, 128, 0, stream>>>(in, Ws[layer], hbuf, DH);
        k_init_agg<<<cdiv((long long)N * DH, 256), 256, 0, stream>>>(
            hbuf, dinv, bs[layer], aggCur, N, DH);
        k_edge_scatter<<<cdiv(E, 256 / DH), 256, 0, stream>>>(
            src, dst, wnorm, hbuf, aggCur, E, DH);
        in = aggCur;             // next layer applies leaky() on load
        float* tmp = aggCur; aggCur = aggNxt; aggNxt = tmp;
    }

    // --- layer 5 (dout = 4, no activation after), aggregate into d_out ---
    float* out = (float*)d_out;
    k_gemm_small<<<cdiv((long long)N * DOUT, 256), 256, 0, stream>>>(
        in, Ws[5], hbuf, N, DH, DOUT);
    k_init_agg<<<cdiv((long long)N * DOUT, 256), 256, 0, stream>>>(
        hbuf, dinv, bs[5], out, N, DOUT);
    k_edge_scatter<<<cdiv(E, 256 / DOUT), 256, 0, stream>>>(
        src, dst, wnorm, hbuf, out, E, DOUT);
}